// EmbedModel_60713657696761
// MI455X (gfx1250) — compile-verified
//
#include <hip/hip_runtime.h>
#include <hip/hip_bf16.h>

#define N_NODES  50000
#define N_EDGES  800000
#define HID      128
#define N_LAYERS 6
#define NUM_GRAPHS 128
#define EMB_W    (HID * (N_LAYERS + 1))   // 896
#define LDS_PITCH 132                     // 132 ≡ 4 (mod 64) -> conflict-free b64 reads

typedef float v2f __attribute__((ext_vector_type(2)));
typedef float v8f __attribute__((ext_vector_type(8)));

// epilogue flags
#define FLAG_RELU       1   // store relu(t)
#define FLAG_ADD_XRES   2   // t += xres[row]; xres[row] = t
#define FLAG_WRITE_XRES 4   // xres[row] = t (no add)
#define FLAG_POOL       8   // pooled[batch[row]*896 + pool_off + col] += stored value

__device__ __forceinline__ void atomic_add_f32(float* p, float v) {
    __hip_atomic_fetch_add(p, v, __ATOMIC_RELAXED, __HIP_MEMORY_SCOPE_AGENT);
}

__device__ __forceinline__ void gemm_epilogue(
    v8f* acc, int m0, int lhalf, int l15, float* Y, float* xres,
    float* pooled, const int* batch, int pool_off, int flags)
{
#pragma unroll
    for (int r = 0; r < 8; ++r) {
        const int row = m0 + lhalf * 8 + r;
        int g = 0;
        if (flags & FLAG_POOL) g = batch[row];
#pragma unroll
        for (int t = 0; t < 8; ++t) {
            const int col = t * 16 + l15;
            float v = acc[t][r];
            if (flags & FLAG_ADD_XRES) {
                v += xres[(size_t)row * HID + col];
                xres[(size_t)row * HID + col] = v;
            } else if (flags & FLAG_WRITE_XRES) {
                xres[(size_t)row * HID + col] = v;
            }
            const float u = (flags & FLAG_RELU) ? fmaxf(v, 0.0f) : v;
            Y[(size_t)row * HID + col] = u;
            if (flags & FLAG_POOL)
                atomic_add_f32(&pooled[(size_t)g * EMB_W + pool_off + col], u);
        }
    }
}

// ---------------------------------------------------------------------------
// Big GEMM (K == 128): W staged transposed in LDS once per workgroup.
// Steady state per k-step: 1 global b64 (A) + 8 ds_load_b64 (B) + 8 WMMA.
// One wave32 per 16-row M-tile, 4 waves per block.
// ---------------------------------------------------------------------------
__global__ void __launch_bounds__(128)
gemm_wmma_lds128(const float* X, const float* __restrict__ W,
                 const float* __restrict__ bias, float* Y,
                 float* xres, float* pooled, const int* __restrict__ batch,
                 int pool_off, int M, int flags)
{
    __shared__ float wt[HID * LDS_PITCH];   // wt[col][k], pitch 132

    // stage W transposed: thread t owns column t; coalesced global reads
#pragma unroll 4
    for (int k = 0; k < HID; ++k)
        wt[(size_t)threadIdx.x * LDS_PITCH + k] = W[(size_t)k * HID + threadIdx.x];
    __syncthreads();

    const int lane  = threadIdx.x & 31;
    const int wave  = threadIdx.x >> 5;
    const int mtile = blockIdx.x * 4 + wave;
    const int mtiles = M >> 4;

    if (mtile < mtiles) {                   // wave-uniform: EXEC all-ones inside
        const int m0    = mtile << 4;
        const int lhalf = lane >> 4;
        const int l15   = lane & 15;
        const float* Xrow = X + (size_t)(m0 + l15) * HID;

        v8f acc[8];
#pragma unroll
        for (int t = 0; t < 8; ++t) {
            float bv = bias[t * 16 + l15];
#pragma unroll
            for (int r = 0; r < 8; ++r) acc[t][r] = bv;
        }

#pragma unroll 4
        for (int kk = 0; kk < HID; kk += 4) {
            const int ka = kk + lhalf * 2;
            v2f a = *(const v2f*)(Xrow + ka);
#pragma unroll
            for (int t = 0; t < 8; ++t) {
                v2f b = *(const v2f*)(&wt[(size_t)(t * 16 + l15) * LDS_PITCH + ka]);
                acc[t] = __builtin_amdgcn_wmma_f32_16x16x4_f32(
                    false, a, false, b, (short)0, acc[t], false, false);
            }
        }
        gemm_epilogue(acc, m0, lhalf, l15, Y, xres, pooled, batch, pool_off, flags);
    }
}

// ---------------------------------------------------------------------------
// Generic-K GEMM (used for the two tiny post-MLP GEMMs, K = 896 / 128).
// ---------------------------------------------------------------------------
__global__ void __launch_bounds__(128)
gemm_wmma_f32(const float* X, const float* __restrict__ W,
              const float* __restrict__ bias, float* Y,
              int M, int K, int flags)
{
    const int lane  = threadIdx.x & 31;
    const int wave  = threadIdx.x >> 5;
    const int mtile = blockIdx.x * 4 + wave;
    const int mtiles = M >> 4;
    if (mtile >= mtiles) return;

    const int m0    = mtile << 4;
    const int lhalf = lane >> 4;
    const int l15   = lane & 15;
    const float* Xrow = X + (size_t)(m0 + l15) * K;

    v8f acc[8];
#pragma unroll
    for (int t = 0; t < 8; ++t) {
        float bv = bias[t * 16 + l15];
#pragma unroll
        for (int r = 0; r < 8; ++r) acc[t][r] = bv;
    }

    for (int kk = 0; kk < K; kk += 4) {
        const int ka = kk + lhalf * 2;
        v2f a = *(const v2f*)(Xrow + ka);
#pragma unroll
        for (int t = 0; t < 8; ++t) {
            const float* Wp = W + (size_t)ka * HID + t * 16 + l15;
            v2f b;
            b.x = Wp[0];
            b.y = Wp[HID];
            acc[t] = __builtin_amdgcn_wmma_f32_16x16x4_f32(
                false, a, false, b, (short)0, acc[t], false, false);
        }
    }
    gemm_epilogue(acc, m0, lhalf, l15, Y, nullptr, nullptr, nullptr, 0, flags);
}

// ---------------------------------------------------------------------------
// Edge gather + scatter-add: agg[dst] += h[src]  (agg pre-initialized to h).
// One wave per edge; indices go through SGPRs (scalar loads), gather is one
// coalesced 512B float4 read per wave, scatter is native L2 f32 atomics.
// ---------------------------------------------------------------------------
__global__ void __launch_bounds__(256)
edge_scatter(const float* __restrict__ h, float* __restrict__ agg,
             const int* __restrict__ src, const int* __restrict__ dst,
             int nEdges)
{
    int e = blockIdx.x * 8 + (threadIdx.x >> 5);
    if (e >= nEdges) return;
    e = __builtin_amdgcn_readfirstlane(e);      // force SGPR -> s_load indices
    const int s = src[e];
    const int d = dst[e];
    const int c = (threadIdx.x & 31) * 4;
    const float4 v = *(const float4*)(h + (size_t)s * HID + c);
    float* p = agg + (size_t)d * HID + c;
    atomic_add_f32(p + 0, v.x);
    atomic_add_f32(p + 1, v.y);
    atomic_add_f32(p + 2, v.z);
    atomic_add_f32(p + 3, v.w);
}

__global__ void __launch_bounds__(256)
copy_f4(const float4* __restrict__ s, float4* __restrict__ d, int n4)
{
    const int i = blockIdx.x * blockDim.x + threadIdx.x;
    if (i < n4) d[i] = s[i];
}

__global__ void __launch_bounds__(256)
zero_f32(float* __restrict__ p, int n)
{
    const int i = blockIdx.x * blockDim.x + threadIdx.x;
    if (i < n) p[i] = 0.0f;
}

// ---------------------------------------------------------------------------
extern "C" void kernel_launch(void* const* d_in, const int* in_sizes, int n_in,
                              void* d_out, int out_size, void* d_ws, size_t ws_size,
                              hipStream_t stream)
{
    const float* x        = (const float*)d_in[0];
    const int*   eidx     = (const int*)  d_in[1];
    const int*   batch    = (const int*)  d_in[2];
    const float* pre_w    = (const float*)d_in[3];
    const float* pre_b    = (const float*)d_in[4];
    const float* conv_w1  = (const float*)d_in[5];
    const float* conv_b1  = (const float*)d_in[6];
    const float* conv_w2  = (const float*)d_in[7];
    const float* conv_b2  = (const float*)d_in[8];
    const float* post_w1  = (const float*)d_in[9];
    const float* post_b1  = (const float*)d_in[10];
    const float* post_w2  = (const float*)d_in[11];
    const float* post_b2  = (const float*)d_in[12];

    const int nEdges = in_sizes[1] / 2;
    const int* src = eidx;
    const int* dst = eidx + nEdges;

    // workspace layout (floats)
    float* h      = (float*)d_ws;                        // N x 128
    float* agg    = h    + (size_t)N_NODES * HID;        // N x 128
    float* xres   = agg  + (size_t)N_NODES * HID;        // N x 128
    float* pooled = xres + (size_t)N_NODES * HID;        // G x 896
    float* tmp    = pooled + (size_t)NUM_GRAPHS * EMB_W; // G x 128

    const int gemmBlocksN = ((N_NODES / 16) + 3) / 4;    // 782
    const int gemmBlocksG = ((NUM_GRAPHS / 16) + 3) / 4; // 2
    const int n4    = N_NODES * HID / 4;
    const int poolN = NUM_GRAPHS * EMB_W;

    zero_f32<<<(poolN + 255) / 256, 256, 0, stream>>>(pooled, poolN);

    // pre: h = x @ pre_w + pre_b ; xres = h ; pooled[:,0:128] += h
    gemm_wmma_lds128<<<gemmBlocksN, 128, 0, stream>>>(
        x, pre_w, pre_b, h, xres, pooled, batch,
        /*pool_off=*/0, N_NODES, FLAG_WRITE_XRES | FLAG_POOL);

    for (int layer = 0; layer < N_LAYERS; ++layer) {
        // agg = h (self term), then agg[dst] += h[src]
        copy_f4<<<(n4 + 255) / 256, 256, 0, stream>>>(
            (const float4*)h, (float4*)agg, n4);
        edge_scatter<<<(nEdges + 7) / 8, 256, 0, stream>>>(
            h, agg, src, dst, nEdges);
        // agg = relu(agg @ w1 + b1)   (in-place, row-disjoint per wave)
        gemm_wmma_lds128<<<gemmBlocksN, 128, 0, stream>>>(
            agg, conv_w1 + (size_t)layer * HID * HID, conv_b1 + layer * HID,
            agg, nullptr, nullptr, nullptr, 0, N_NODES, FLAG_RELU);
        // z = agg @ w2 + b2 ; [odd: z += xres, xres = z] ; h = relu(z) ; pool
        int flags = FLAG_RELU | FLAG_POOL | ((layer & 1) ? FLAG_ADD_XRES : 0);
        gemm_wmma_lds128<<<gemmBlocksN, 128, 0, stream>>>(
            agg, conv_w2 + (size_t)layer * HID * HID, conv_b2 + layer * HID,
            h, xres, pooled, batch, (layer + 1) * HID, N_NODES, flags);
    }

    // tmp = relu(pooled @ post_w1 + post_b1)   [128 x 896] @ [896 x 128]
    gemm_wmma_f32<<<gemmBlocksG, 128, 0, stream>>>(
        pooled, post_w1, post_b1, tmp, NUM_GRAPHS, EMB_W, FLAG_RELU);
    // out = tmp @ post_w2 + post_b2
    gemm_wmma_f32<<<gemmBlocksG, 128, 0, stream>>>(
        tmp, post_w2, post_b2, (float*)d_out, NUM_GRAPHS, HID, 0);
}